// RestrictedSpectralOperator_70394513981642
// MI455X (gfx1250) — compile-verified
//
#include <hip/hip_runtime.h>
#include <hip/hip_bf16.h>
#include <cstddef>

// ---------------------------------------------------------------------------
// RestrictedSpectralOperator on MI455X (gfx1250, wave32, WMMA f32 16x16x4)
//
// y[b,t,h,w,c] = scalar * irfft2( mask * rfft2(x transposed) )
// Implemented as three truncated-DFT GEMM stages (see header analysis):
//   K1: Zf[bt][reim][h][kw][c]  = A1[32,128] . X[bt,h][128w,32c]
//   K2: Z2[bt][i][col]          = Q[256,256] . Zf[bt][256][512]
//   K3: y[bt,h][128w,32c]       = s * A3[128,32] . Z2view[bt,h][32m,32c]
// All GEMMs use v_wmma_f32_16x16x4_f32 (full fp32 precision).
// ---------------------------------------------------------------------------

typedef __attribute__((ext_vector_type(2))) float v2f;
typedef __attribute__((ext_vector_type(8))) float v8f;

__device__ __forceinline__ v8f wmma4(v2f a, v2f b, v8f c) {
  // D = A(16x4) * B(4x16) + C, f32 everywhere.
  return __builtin_amdgcn_wmma_f32_16x16x4_f32(
      /*neg_a=*/false, a, /*neg_b=*/false, b,
      /*c_mod=*/(short)0, c, /*reuse_a=*/false, /*reuse_b=*/false);
}

// Workspace layout (floats):
//   Q  : [256][256]                      @ 0        (65536)
//   A1 : [32][128]  W-forward DFT rows   @ 65536    (4096)
//   A3 : [128][32]  W-inverse DFT rows   @ 69632    (4096)
//   Zf : [64][256][512]                  @ 73728    (8388608)
//   Z2 : [64][256][512]                  @ 8462336  (8388608)
// Total 16,850,944 floats = 67.4 MB (fits in the 192MB L2 alongside i/o).

__global__ __launch_bounds__(256) void fno_tables(float* __restrict__ Q,
                                                  float* __restrict__ A1,
                                                  float* __restrict__ A3) {
  int t = blockIdx.x * 256 + threadIdx.x;
  if (t < 65536) {
    // Q = [[Pr, -Pi], [Pi, Pr]], circulant in (h' - h) mod 128.
    int i = t >> 8, j = t & 255;
    int ri = i >> 7, h1 = i & 127;
    int rj = j >> 7, h0 = j & 127;
    int d = (h1 - h0) & 127;
    float pr = 1.0f;
#pragma unroll
    for (int k = 1; k <= 15; ++k) pr += 2.0f * cospif((float)(k * d) * (1.0f / 64.0f));
    pr += cospif((float)(16 * d) * (1.0f / 64.0f));
    pr *= (1.0f / 128.0f);
    float pim = -sinpif((float)(16 * d) * (1.0f / 64.0f)) * (1.0f / 128.0f);
    float q;
    if (ri == rj)      q = pr;
    else if (ri == 0)  q = -pim;  // top-right block: -Pi
    else               q = pim;   // bottom-left block: +Pi
    Q[t] = q;
  } else if (t < 69632) {
    // A1[m][w]: m<16 -> cos(2*pi*m*w/128) ; m>=16 -> -sin(2*pi*(m-16)*w/128)
    int t2 = t - 65536;
    int m = t2 >> 7, w = t2 & 127;
    int k = m & 15;
    float arg = (float)(k * w) * (1.0f / 64.0f);
    A1[t2] = (m < 16) ? cospif(arg) : -sinpif(arg);
  } else if (t < 73728) {
    // A3[w][m]: irfft reconstruction weights for the 16 kept kw modes.
    int t3 = t - 69632;
    int w = t3 >> 5, m = t3 & 31;
    int k = m & 15;
    float arg = (float)(k * w) * (1.0f / 64.0f);
    float val;
    if (m == 0)       val = 1.0f / 128.0f;
    else if (m < 16)  val = cospif(arg) * (2.0f / 128.0f);
    else if (m == 16) val = 0.0f;                       // imag of kw=0 dropped by irfft
    else              val = -sinpif(arg) * (2.0f / 128.0f);
    A3[t3] = val;
  }
}

// K1: one wave per (bt,h) site; 32x32 output tile, K=128 over w.
__global__ __launch_bounds__(32) void k1_wfwd(const float* __restrict__ x,
                                              const float* __restrict__ A1,
                                              float* __restrict__ Zf) {
  const int lane = threadIdx.x;
  const int G = lane >> 4, lm = lane & 15;
  const int site = blockIdx.x;          // 0..8191 = bt*128 + h
  const int bt = site >> 7, h = site & 127;
  const float* __restrict__ X = x + (size_t)site * (128 * 32);

  v8f acc00 = {}, acc01 = {}, acc10 = {}, acc11 = {};
#pragma unroll 4
  for (int k0 = 0; k0 < 128; k0 += 4) {
    const int kk = k0 + 2 * G;
    v2f a0 = *(const v2f*)(A1 + (size_t)lm * 128 + kk);
    v2f a1 = *(const v2f*)(A1 + (size_t)(16 + lm) * 128 + kk);
    v2f b0, b1;
    b0.x = X[(size_t)kk * 32 + lm];
    b0.y = X[(size_t)(kk + 1) * 32 + lm];
    b1.x = X[(size_t)kk * 32 + 16 + lm];
    b1.y = X[(size_t)(kk + 1) * 32 + 16 + lm];
    acc00 = wmma4(a0, b0, acc00);
    acc01 = wmma4(a0, b1, acc01);
    acc10 = wmma4(a1, b0, acc10);
    acc11 = wmma4(a1, b1, acc11);
  }
  // Store Zf[bt][reim*128+h][kw*32+c]; reim = mt, kw = v + 8*G.
  float* __restrict__ Zbt = Zf + (size_t)bt * 131072;
#pragma unroll
  for (int v = 0; v < 8; ++v) {
    const int kw = v + 8 * G;
    Zbt[(size_t)h * 512 + kw * 32 + lm]                = acc00[v];
    Zbt[(size_t)h * 512 + kw * 32 + 16 + lm]           = acc01[v];
    Zbt[(size_t)(128 + h) * 512 + kw * 32 + lm]        = acc10[v];
    Zbt[(size_t)(128 + h) * 512 + kw * 32 + 16 + lm]   = acc11[v];
  }
}

// K2: complex circulant H-projection as one real GEMM [256,256]x[256,512] per bt.
__global__ __launch_bounds__(32) void k2_hop(const float* __restrict__ Q,
                                             const float* __restrict__ Zf,
                                             float* __restrict__ Z2) {
  const int lane = threadIdx.x;
  const int G = lane >> 4, lm = lane & 15;
  const int id = blockIdx.x;            // 64 bt * 8 itile * 16 coltile = 8192
  const int bt = id >> 7;
  const int r = id & 127;
  const int i0 = (r >> 4) * 32;
  const int col0 = (r & 15) * 32;
  const float* __restrict__ Zbt = Zf + (size_t)bt * 131072;
  float* __restrict__ Obt = Z2 + (size_t)bt * 131072;

  v8f acc00 = {}, acc01 = {}, acc10 = {}, acc11 = {};
#pragma unroll 4
  for (int j0 = 0; j0 < 256; j0 += 4) {
    const int kk = j0 + 2 * G;
    v2f a0 = *(const v2f*)(Q + (size_t)(i0 + lm) * 256 + kk);
    v2f a1 = *(const v2f*)(Q + (size_t)(i0 + 16 + lm) * 256 + kk);
    v2f b0, b1;
    b0.x = Zbt[(size_t)kk * 512 + col0 + lm];
    b0.y = Zbt[(size_t)(kk + 1) * 512 + col0 + lm];
    b1.x = Zbt[(size_t)kk * 512 + col0 + 16 + lm];
    b1.y = Zbt[(size_t)(kk + 1) * 512 + col0 + 16 + lm];
    acc00 = wmma4(a0, b0, acc00);
    acc01 = wmma4(a0, b1, acc01);
    acc10 = wmma4(a1, b0, acc10);
    acc11 = wmma4(a1, b1, acc11);
  }
#pragma unroll
  for (int v = 0; v < 8; ++v) {
    const int m = v + 8 * G;
    Obt[(size_t)(i0 + m) * 512 + col0 + lm]           = acc00[v];
    Obt[(size_t)(i0 + m) * 512 + col0 + 16 + lm]      = acc01[v];
    Obt[(size_t)(i0 + 16 + m) * 512 + col0 + lm]      = acc10[v];
    Obt[(size_t)(i0 + 16 + m) * 512 + col0 + 16 + lm] = acc11[v];
  }
}

// K3: W-inverse + scalar. One wave per (site, w-quarter); K=32.
__global__ __launch_bounds__(32) void k3_winv(const float* __restrict__ A3,
                                              const float* __restrict__ Z2,
                                              const float* __restrict__ scalar,
                                              float* __restrict__ y) {
  const int lane = threadIdx.x;
  const int G = lane >> 4, lm = lane & 15;
  const int id = blockIdx.x;            // 8192 sites * 4 = 32768
  const int site = id >> 2;
  const int w0 = (id & 3) * 32;
  const int bt = site >> 7, h = site & 127;
  const float* __restrict__ Zbt = Z2 + (size_t)bt * 131072;
  const float s = scalar[0];

  v8f acc00 = {}, acc01 = {}, acc10 = {}, acc11 = {};
#pragma unroll
  for (int k0 = 0; k0 < 32; k0 += 4) {
    const int kk = k0 + 2 * G;          // even, 0..30
    const int re0 = kk >> 4, kw0 = kk & 15;
    const int re1 = (kk + 1) >> 4, kw1 = (kk + 1) & 15;
    v2f a0 = *(const v2f*)(A3 + (size_t)(w0 + lm) * 32 + kk);
    v2f a1 = *(const v2f*)(A3 + (size_t)(w0 + 16 + lm) * 32 + kk);
    v2f b0, b1;
    b0.x = Zbt[(size_t)(re0 * 128 + h) * 512 + kw0 * 32 + lm];
    b0.y = Zbt[(size_t)(re1 * 128 + h) * 512 + kw1 * 32 + lm];
    b1.x = Zbt[(size_t)(re0 * 128 + h) * 512 + kw0 * 32 + 16 + lm];
    b1.y = Zbt[(size_t)(re1 * 128 + h) * 512 + kw1 * 32 + 16 + lm];
    acc00 = wmma4(a0, b0, acc00);
    acc01 = wmma4(a0, b1, acc01);
    acc10 = wmma4(a1, b0, acc10);
    acc11 = wmma4(a1, b1, acc11);
  }
  float* __restrict__ Y = y + (size_t)site * (128 * 32);
#pragma unroll
  for (int v = 0; v < 8; ++v) {
    const int wv = v + 8 * G;
    Y[(size_t)(w0 + wv) * 32 + lm]           = s * acc00[v];
    Y[(size_t)(w0 + wv) * 32 + 16 + lm]      = s * acc01[v];
    Y[(size_t)(w0 + 16 + wv) * 32 + lm]      = s * acc10[v];
    Y[(size_t)(w0 + 16 + wv) * 32 + 16 + lm] = s * acc11[v];
  }
}

extern "C" void kernel_launch(void* const* d_in, const int* in_sizes, int n_in,
                              void* d_out, int out_size, void* d_ws, size_t ws_size,
                              hipStream_t stream) {
  (void)in_sizes; (void)n_in; (void)out_size; (void)ws_size;
  const float* x = (const float*)d_in[0];
  const float* scalar = (const float*)d_in[1];
  float* y = (float*)d_out;

  float* wsf = (float*)d_ws;
  float* Q  = wsf;
  float* A1 = wsf + 65536;
  float* A3 = wsf + 69632;
  float* Zf = wsf + 73728;
  float* Z2 = Zf + 8388608;

  fno_tables<<<288, 256, 0, stream>>>(Q, A1, A3);
  k1_wfwd<<<8192, 32, 0, stream>>>(x, A1, Zf);
  k2_hop<<<8192, 32, 0, stream>>>(Q, Zf, Z2);
  k3_winv<<<32768, 32, 0, stream>>>(A3, Z2, scalar, y);
}